// Fdnet3D_35278861369494
// MI455X (gfx1250) — compile-verified
//
#include <hip/hip_runtime.h>

#define H 128
#define W 128
#define NIMG 64          // BS*CH images, channels folded into batch (Cin=1)
#define F 16
#define LSTRIDE 132      // padded LDS row stride (floats): conflict-free WMMA loads

typedef __attribute__((ext_vector_type(2))) float v2f;
typedef __attribute__((ext_vector_type(4))) float v4f;
typedef __attribute__((ext_vector_type(8))) float v8f;

// ---------------------------------------------------------------------------
// Kernel 1: collapse the conv stack into banded coefficient matrices.
//   ATmat[c][m] : coeff of x[row,m] in the horizontal deviation at column c
//   Bmat [r][m] : coeff of x[m,col] in the vertical   deviation at row r
// Band is |m-j| <= 2 everywhere (boundary cases included).
// ---------------------------------------------------------------------------
__device__ void build_row(float* __restrict__ rowOut, int j,
                          const float* __restrict__ w1,  const float* __restrict__ w1L,
                          const float* __restrict__ w1R, const float* __restrict__ w2,
                          const float* __restrict__ w2L, const float* __restrict__ w2R,
                          const float* __restrict__ vfirst, const float* __restrict__ usec)
{
  float band[5] = {0.f, 0.f, 0.f, 0.f, 0.f};   // coeffs of x[j-2 .. j+2]
  const int base = j - 2;

  // expand first-layer stencil taps of d-field position m, scaled by w2s
  auto addD = [&](int m, float w2s, int f) {
    if (m == 0) {
      band[0 - base]     += w2s * w1L[f*2+0];
      band[1 - base]     += w2s * w1L[f*2+1];
    } else if (m == W-1) {
      band[(W-2) - base] += w2s * w1R[f*2+0];
      band[(W-1) - base] += w2s * w1R[f*2+1];
    } else {
      band[(m-1) - base] += w2s * w1[f*3+0];
      band[(m  ) - base] += w2s * w1[f*3+1];
      band[(m+1) - base] += w2s * w1[f*3+2];
    }
  };

  // ---- first-order contribution: sum_f vfirst[f] * d1[f] ----
  if (j == 0) {
    for (int f = 0; f < F; ++f) {
      band[2] += vfirst[f] * w1L[f*2+0];
      band[3] += vfirst[f] * w1L[f*2+1];
    }
  } else if (j == W-1) {
    for (int f = 0; f < F; ++f) {
      band[1] += vfirst[f] * w1R[f*2+0];
      band[2] += vfirst[f] * w1R[f*2+1];
    }
  } else {
    for (int f = 0; f < F; ++f)
      for (int k = 0; k < 3; ++k)
        band[1 + k] += vfirst[f] * w1[f*3+k];
  }

  // ---- second-order: sum_g usec[g] * d2[g], d2 = conv(d1, w2) ----
  for (int f = 0; f < F; ++f) {
    float b0=0.f, b1=0.f, b2=0.f, bl0=0.f, bl1=0.f, br0=0.f, br1=0.f;
    for (int g = 0; g < F; ++g) {
      const float u = usec[g];
      b0  += u * w2 [g*F*3 + f*3 + 0];
      b1  += u * w2 [g*F*3 + f*3 + 1];
      b2  += u * w2 [g*F*3 + f*3 + 2];
      bl0 += u * w2L[g*F*2 + f*2 + 0];
      bl1 += u * w2L[g*F*2 + f*2 + 1];
      br0 += u * w2R[g*F*2 + f*2 + 0];
      br1 += u * w2R[g*F*2 + f*2 + 1];
    }
    if (j == 0)        { addD(0,   bl0, f); addD(1,   bl1, f); }
    else if (j == W-1) { addD(W-2, br0, f); addD(W-1, br1, f); }
    else               { addD(j-1, b0,  f); addD(j,   b1,  f); addD(j+1, b2, f); }
  }

  for (int m = 0; m < W; ++m) {
    const int t = m - base;
    rowOut[m] = (t >= 0 && t < 5) ? band[t] : 0.f;
  }
}

__global__ __launch_bounds__(128) void build_mats(
    const float* wX1, const float* wX1L, const float* wX1R,
    const float* wY1, const float* wY1L, const float* wY1R,
    const float* wX2, const float* wX2L, const float* wX2R,
    const float* wY2, const float* wY2L, const float* wY2R,
    const float* wfc, float* ATmat, float* Bmat)
{
  const int j = threadIdx.x;
  if (j >= W) return;
  // combine channel order: [dX | dY | dX2 | dY2] -> wfc slices 0,F,2F,3F
  build_row(ATmat + j*W, j, wX1, wX1L, wX1R, wX2, wX2L, wX2R, wfc + 0, wfc + 2*F);
  build_row(Bmat  + j*W, j, wY1, wY1L, wY1R, wY2, wY2L, wY2R, wfc + F, wfc + 3*F);
}

// ---------------------------------------------------------------------------
// Kernel 2: fdblock in-LDS iterations of  x <- x + X*Acoef + Bcoef*X  per
// image, banded K-range only, via V_WMMA_F32_16X16X4_F32.
// One workgroup (8 wave32) per image; wave w owns tile-row w (16 rows).
// All iteration-invariant coefficient operands are hoisted into VGPRs, so the
// hot loop is pure ds_load + v_wmma + ds_store.
// ---------------------------------------------------------------------------
__global__ __launch_bounds__(256, 1) void fd_iter(
    const float* __restrict__ xIn, const float* __restrict__ ATmat,
    const float* __restrict__ Bmat, const int* __restrict__ fdblockPtr,
    float* __restrict__ xOut)
{
  extern __shared__ float lds[];
  float* buf0 = lds;
  float* buf1 = lds + H * LSTRIDE;

  const int tid = threadIdx.x;
  const int img = blockIdx.x;
  const float* gx = xIn  + (size_t)img * (H * W);
  float*       gy = xOut + (size_t)img * (H * W);

  const int lane = tid & 31;
  const int rL   = lane & 15;    // N / M sub-index within tile
  const int hi   = lane >> 4;    // half-wave selector (K pair / row-half)
  const int it   = tid >> 5;     // tile row owned by this wave

  // pull coefficient matrices toward the caches (global_prefetch_b8)
  __builtin_prefetch(&ATmat[tid * 64], 0, 1);
  __builtin_prefetch(&Bmat[tid * 64], 0, 1);

  // ---- hoist iteration-invariant WMMA coefficient operands into VGPRs ----
  // horizontal B-operands: Acoef[m][c] = ATmat[c][m]; K-blocks kb in [4jt-1, 4jt+4]
  v2f hcoef[48];
#pragma unroll
  for (int jt = 0; jt < 8; ++jt) {
#pragma unroll
    for (int s = 0; s < 6; ++s) {
      const int kb = 4 * jt - 1 + s;            // compile-time
      if (kb >= 0 && kb <= 31) {
        hcoef[jt * 6 + s] = *(const v2f*)&ATmat[(16 * jt + rL) * W + 4 * kb + 2 * hi];
      } else {
        v2f z = {0.f, 0.f};
        hcoef[jt * 6 + s] = z;                  // band edge: dead slot
      }
    }
  }
  // vertical A-operands: Bmat[r][m]; K-blocks kb in [4it-1, 4it+4] (it uniform)
  v2f vcoef[6];
  int vmb[6];
#pragma unroll
  for (int s = 0; s < 6; ++s) {
    const int kb  = 4 * it - 1 + s;
    const int kbc = kb < 0 ? 0 : (kb > 31 ? 31 : kb);
    vmb[s] = 4 * kbc + 2 * hi;
    v2f c = *(const v2f*)&Bmat[(16 * it + rL) * W + vmb[s]];
    if (kb < 0 || kb > 31) { c.x = 0.f; c.y = 0.f; }   // wave-uniform edge
    vcoef[s] = c;
  }

  // ---- load image -> buf0 (b128; LSTRIDE%4==0 keeps 16B alignment) ----
  for (int i4 = tid; i4 < (H * W) / 4; i4 += 256) {
    const int lin = i4 * 4;
    const int r = lin >> 7, c = lin & 127;
    *(v4f*)&buf0[r * LSTRIDE + c] = *(const v4f*)&gx[lin];
  }
  __syncthreads();

  const int iters = *fdblockPtr;
  const int hrow  = (16 * it + rL) * LSTRIDE + 2 * hi;   // horizontal A-op row base
  const int crow  = (16 * it + 8 * hi) * LSTRIDE;        // C/D tile row base

  for (int t = 0; t < iters; ++t) {
    const float* src = (t & 1) ? buf1 : buf0;
    float*       dst = (t & 1) ? buf0 : buf1;

#pragma unroll
    for (int jt = 0; jt < 8; ++jt) {
      const int col = 16 * jt + rL;

      // C init = current x tile (C/D layout: M = r + 8*hi, N = rL)
      v8f acc;
#pragma unroll
      for (int r = 0; r < 8; ++r)
        acc[r] = src[crow + r * LSTRIDE + col];

      // horizontal: acc += Xtile(16xK) * Acoef(Kx16)
#pragma unroll
      for (int s = 0; s < 6; ++s) {
        const int kb  = 4 * jt - 1 + s;                  // compile-time
        const int kbc = kb < 0 ? 0 : (kb > 31 ? 31 : kb);
        v2f aop = *(const v2f*)&src[hrow + 4 * kbc];     // folds into ds offset
        acc = __builtin_amdgcn_wmma_f32_16x16x4_f32(false, aop, false,
                                                    hcoef[jt * 6 + s],
                                                    (short)0, acc, false, false);
      }

      // vertical: acc += Bcoef(16xK) * Xslice(Kx16)
#pragma unroll
      for (int s = 0; s < 6; ++s) {
        v2f bop;
        bop.x = src[(vmb[s]    ) * LSTRIDE + col];
        bop.y = src[(vmb[s] + 1) * LSTRIDE + col];
        acc = __builtin_amdgcn_wmma_f32_16x16x4_f32(false, vcoef[s], false, bop,
                                                    (short)0, acc, false, false);
      }

#pragma unroll
      for (int r = 0; r < 8; ++r)
        dst[crow + r * LSTRIDE + col] = acc[r];
    }
    __syncthreads();
  }

  const float* res = (iters & 1) ? buf1 : buf0;
  for (int i4 = tid; i4 < (H * W) / 4; i4 += 256) {
    const int lin = i4 * 4;
    const int r = lin >> 7, c = lin & 127;
    *(v4f*)&gy[lin] = *(const v4f*)&res[r * LSTRIDE + c];
  }
}

// ---------------------------------------------------------------------------
extern "C" void kernel_launch(void* const* d_in, const int* in_sizes, int n_in,
                              void* d_out, int out_size, void* d_ws, size_t ws_size,
                              hipStream_t stream)
{
  (void)in_sizes; (void)n_in; (void)out_size; (void)ws_size;
  const float* xInput = (const float*)d_in[0];
  const float* wX1  = (const float*)d_in[1];
  const float* wX1L = (const float*)d_in[2];
  const float* wX1R = (const float*)d_in[3];
  const float* wY1  = (const float*)d_in[4];
  const float* wY1L = (const float*)d_in[5];
  const float* wY1R = (const float*)d_in[6];
  const float* wX2  = (const float*)d_in[7];
  const float* wX2L = (const float*)d_in[8];
  const float* wX2R = (const float*)d_in[9];
  const float* wY2  = (const float*)d_in[10];
  const float* wY2L = (const float*)d_in[11];
  const float* wY2R = (const float*)d_in[12];
  const float* wfc  = (const float*)d_in[13];
  const int*   fdb  = (const int*)d_in[14];

  float* ATmat = (float*)d_ws;           // 128*128 f32
  float* Bmat  = ATmat + H * W;          // 128*128 f32

  build_mats<<<1, 128, 0, stream>>>(wX1, wX1L, wX1R, wY1, wY1L, wY1R,
                                    wX2, wX2L, wX2R, wY2, wY2L, wY2R,
                                    wfc, ATmat, Bmat);

  const size_t ldsBytes = (size_t)2 * H * LSTRIDE * sizeof(float);  // 135168 B
  fd_iter<<<NIMG, 256, ldsBytes, stream>>>(xInput, ATmat, Bmat, fdb, (float*)d_out);
}